// GaussianDecoupler_77412490543575
// MI455X (gfx1250) — compile-verified
//
#include <hip/hip_runtime.h>
#include <math.h>

typedef __attribute__((ext_vector_type(2))) float v2f;
typedef __attribute__((ext_vector_type(8))) float v8f;

#define B_    32
#define C_    256
#define HW_   56
#define PLANE (HW_*HW_)
#define KS    31
#define PAD   15
#define TILE  16
#define XT    48   // K-padded halo tile extent (46 valid rows/cols, rows 46/47 zero)
#define XP    49   // LDS row stride for X (odd -> conflict-free)
#define HP    17   // LDS row stride for H
#define VKOFF 16   // tap j lives at vkZ[m*64 + VKOFF + j]; rest is zero padding

// ---------------- Stage 1: global average pool, one wave per (b,c) plane ----
__global__ __launch_bounds__(32) void gd_gap(const float* __restrict__ Fs,
                                             float* __restrict__ gap) {
    int bc = blockIdx.x;
    const float* p = Fs + (size_t)bc * PLANE;
    int l = threadIdx.x;
    float s = 0.f;
    for (int i = l; i < PLANE; i += 32) s += p[i];
    for (int m = 16; m >= 1; m >>= 1) s += __shfl_xor(s, m, 32);
    if (l == 0) gap[bc] = s * (1.0f / (float)PLANE);
}

// ---------------- Stage 2: selector + separable 1D kernels (1 block) --------
__global__ __launch_bounds__(32) void gd_sel(const float* __restrict__ gap,
    const float* __restrict__ sel_w, const float* __restrict__ gamma,
    const float* __restrict__ beta,  const float* __restrict__ mean,
    const float* __restrict__ var,   const float* __restrict__ gb,
    float* __restrict__ vk, float* __restrict__ wmix) {
    int l = threadIdx.x;
    // 1D kernel v_m[j] = G_m[15,j] / sum_j G_m[15,j]  (G_m = v_m v_m^T)
    for (int m = 0; m < 3; ++m) {
        float rs = 0.f;
        for (int j = 0; j < KS; ++j) rs += gb[m*KS*KS + PAD*KS + j];
        if (l < KS)  vk[m*32 + l]  = gb[m*KS*KS + PAD*KS + l] / rs;
        if (l == KS) vk[m*32 + 31] = 0.f;                 // tap 31 = 0 pad
    }
    // total sum of each stored basis (for faithful renormalization)
    float S[3];
    for (int m = 0; m < 3; ++m) {
        float s = 0.f;
        for (int t = l; t < KS*KS; t += 32) s += gb[m*KS*KS + t];
        for (int k = 16; k >= 1; k >>= 1) s += __shfl_xor(s, k, 32);
        S[m] = s;
    }
    // per-sample: logits -> BN(eval) -> softmax -> renormalized mix weights
    int b = l;                                            // B_ == 32 lanes
    float lg[3];
    for (int m = 0; m < 3; ++m) {
        float d = 0.f;
        for (int c = 0; c < C_; ++c) d += gap[b*C_ + c] * sel_w[m*C_ + c];
        lg[m] = (d - mean[m]) * rsqrtf(var[m] + 1e-5f) * gamma[m] + beta[m];
    }
    float mx = fmaxf(lg[0], fmaxf(lg[1], lg[2]));
    float e[3], se = 0.f;
    for (int m = 0; m < 3; ++m) { e[m] = expf(lg[m] - mx); se += e[m]; }
    float ks = 0.f;
    for (int m = 0; m < 3; ++m) { e[m] /= se; ks += e[m] * S[m]; }
    float inv = 1.f / (ks + 1e-12f);
    for (int m = 0; m < 3; ++m) wmix[b*3 + m] = e[m] * inv;
}

// ---------------- Stage 3: fused separable conv via f32 WMMA ----------------
// One wave per 16x16 output tile. Horizontal: H_m(48x16) = X(48x48)*T_m with
// T_m a banded Toeplitz of v_m, fragments gathered branch-free from a zero-
// padded LDS copy of v_m. Vertical: Fl = sum_m (w_m*Tv_m)(16x48)*H_m into one
// v8f accumulator. Fh = Fs - Fl straight from the LDS halo tile.
__global__ __launch_bounds__(32) void gd_conv(const float* __restrict__ Fs,
    const float* __restrict__ vk, const float* __restrict__ wmix,
    float* __restrict__ Fl, float* __restrict__ Fh) {
    __shared__ float Xs[XT * XP];
    __shared__ float Hs[3 * XT * HP];
    __shared__ float vkZ[3 * 64];       // zero-padded taps: [VKOFF..VKOFF+30]

    const int l = threadIdx.x;
    const int half = l >> 4;            // K half-split across lane halves
    const int l15  = l & 15;
    const int tile = blockIdx.x;        // 0..15
    const int ty0 = (tile >> 2) * TILE, tx0 = (tile & 3) * TILE;
    const int c = blockIdx.y, b = blockIdx.z;
    const float* plane = Fs + ((size_t)b * C_ + c) * PLANE;

    for (int i = l; i < 3 * 64; i += 32) vkZ[i] = 0.f;
    for (int i = l; i < XT * XP; i += 32) Xs[i] = 0.f;
    __syncthreads();
    // taps (vk[m*32+31] is already 0, lands harmlessly in the padding)
    {
        int m = l >> 4, j = (l & 15);               // 32 lanes cover 2 rows
        vkZ[m * 64 + VKOFF + j]      = vk[m * 32 + j];
        vkZ[m * 64 + VKOFF + j + 16] = vk[m * 32 + j + 16];
        if (l < 32) vkZ[2 * 64 + VKOFF + l] = vk[2 * 32 + l];
    }

    // load 46x46 halo region (zeros outside the image)
    for (int r = 0; r < 46; ++r) {
        int gy = ty0 - PAD + r;
        if (gy < 0 || gy >= HW_) continue;
        int gx = tx0 - PAD + l;
        if (gx >= 0 && gx < HW_) Xs[r * XP + l] = plane[gy * HW_ + gx];
        int c2 = l + 32, gx2 = gx + 32;
        if (c2 < 46 && gx2 >= 0 && gx2 < HW_) Xs[r * XP + c2] = plane[gy * HW_ + gx2];
    }
    __syncthreads();

    // per-lane base into the padded tap table: idx = VKOFF + (kb - l15)
    const int tbase = VKOFF + 2 * half - l15;       // + 4*kc per chunk

    // ---- horizontal pass: 3 row-groups x 3 bases x 12 K-chunks of WMMA ----
    for (int g = 0; g < 3; ++g) {
        float a0[12], a1[12];
#pragma unroll
        for (int kc = 0; kc < 12; ++kc) {           // A: row = l15, K by half
            int kb = 4 * kc + 2 * half;
            a0[kc] = Xs[(16 * g + l15) * XP + kb];
            a1[kc] = Xs[(16 * g + l15) * XP + kb + 1];
        }
#pragma unroll
        for (int m = 0; m < 3; ++m) {
            float b0[12], b1[12];
#pragma unroll
            for (int kc = 0; kc < 12; ++kc) {       // branch-free Toeplitz B
                b0[kc] = vkZ[m * 64 + tbase + 4 * kc];
                b1[kc] = vkZ[m * 64 + tbase + 4 * kc + 1];
            }
            v8f acc = {};
#pragma unroll
            for (int kc = 0; kc < 12; ++kc) {
                v2f A = { a0[kc], a1[kc] };
                v2f Bv = { b0[kc], b1[kc] };
                acc = __builtin_amdgcn_wmma_f32_16x16x4_f32(
                    false, A, false, Bv, (short)0, acc, false, false);
            }
#pragma unroll
            for (int i = 0; i < 8; ++i)             // D: M = i + 8*half, N = l15
                Hs[(m * XT + 16 * g + i + 8 * half) * HP + l15] = acc[i];
        }
    }
    __syncthreads();

    // ---- vertical pass: weighted Toeplitz x H, one accumulator over all m --
    float wmv[3] = { wmix[b * 3 + 0], wmix[b * 3 + 1], wmix[b * 3 + 2] };
    v8f acc = {};
#pragma unroll
    for (int m = 0; m < 3; ++m) {
        float t0[12], t1[12], h0[12], h1[12];
#pragma unroll
        for (int kc = 0; kc < 12; ++kc) {
            int kb = 4 * kc + 2 * half;
            t0[kc] = vkZ[m * 64 + tbase + 4 * kc];          // Tv[y][k]=v[k-y]
            t1[kc] = vkZ[m * 64 + tbase + 4 * kc + 1];
            h0[kc] = Hs[(m * XT + kb)     * HP + l15];
            h1[kc] = Hs[(m * XT + kb + 1) * HP + l15];
        }
#pragma unroll
        for (int kc = 0; kc < 12; ++kc) {
            v2f A  = { wmv[m] * t0[kc], wmv[m] * t1[kc] };
            v2f Bv = { h0[kc], h1[kc] };
            acc = __builtin_amdgcn_wmma_f32_16x16x4_f32(
                false, A, false, Bv, (short)0, acc, false, false);
        }
    }
#pragma unroll
    for (int i = 0; i < 8; ++i) {
        int ly = i + 8 * half, lx = l15;
        int gy = ty0 + ly, gx = tx0 + lx;
        if (gy < HW_ && gx < HW_) {
            float fl = acc[i];
            float fs = Xs[(ly + PAD) * XP + (lx + PAD)];
            size_t o = ((size_t)b * C_ + c) * PLANE + (size_t)gy * HW_ + gx;
            Fl[o] = fl;
            Fh[o] = fs - fl;
        }
    }
}

extern "C" void kernel_launch(void* const* d_in, const int* in_sizes, int n_in,
                              void* d_out, int out_size, void* d_ws, size_t ws_size,
                              hipStream_t stream) {
    const float* Fs    = (const float*)d_in[0];
    const float* sel_w = (const float*)d_in[1];
    const float* gamma = (const float*)d_in[2];
    const float* beta  = (const float*)d_in[3];
    const float* mean  = (const float*)d_in[4];
    const float* var   = (const float*)d_in[5];
    const float* gb    = (const float*)d_in[6];

    float* gap  = (float*)d_ws;          // B*C floats
    float* vk   = gap + B_ * C_;         // 3*32 floats (1D kernels, tap31=0)
    float* wmix = vk + 3 * 32;           // B*3 floats (renormalized weights)

    float* Fl = (float*)d_out;
    float* Fh = Fl + (size_t)B_ * C_ * PLANE;

    gd_gap<<<B_ * C_, 32, 0, stream>>>(Fs, gap);
    gd_sel<<<1, 32, 0, stream>>>(gap, sel_w, gamma, beta, mean, var, gb, vk, wmix);
    dim3 grid(16, C_, B_);               // 4x4 tiles per plane
    gd_conv<<<grid, 32, 0, stream>>>(Fs, vk, wmix, Fl, Fh);
}